// HeteroGraphEncoder_56427280335131
// MI455X (gfx1250) — compile-verified
//
#include <hip/hip_runtime.h>
#include <cstring>

typedef __attribute__((ext_vector_type(2))) float v2f;
typedef __attribute__((ext_vector_type(8))) float v8f;

#define LRELU_SLOPE 0.2f

// ---------------- device helpers ----------------

__device__ __forceinline__ unsigned f2ord(float f) {
  unsigned u = __float_as_uint(f);
  return (u & 0x80000000u) ? ~u : (u | 0x80000000u);
}
__device__ __forceinline__ float ord2f(unsigned u) {
  u = (u & 0x80000000u) ? (u & 0x7FFFFFFFu) : ~u;
  return __uint_as_float(u);
}

// ---------------- utility kernels ----------------

__global__ void fill_u32_kernel(unsigned* __restrict__ p, long long n, unsigned v) {
  long long i = (long long)blockIdx.x * blockDim.x + threadIdx.x;
  if (i < n) p[i] = v;
}

// out[n,k] = b1[k] (+ b2[k])   (accumulator init with biases of the summed relations)
__global__ void init_bias_kernel(float* __restrict__ out, long long n, int K,
                                 const float* __restrict__ b1,
                                 const float* __restrict__ b2) {
  long long i = (long long)blockIdx.x * blockDim.x + threadIdx.x;
  if (i < n) {
    int k = (int)(i % K);
    float v = b1[k];
    if (b2) v += b2[k];
    out[i] = v;
  }
}

// V[f,h] = sum_c W[f, h*C+c] * avec[h*C+c]   (fold attention vector through W)
__global__ void vfold_kernel(const float* __restrict__ W, const float* __restrict__ avec,
                             float* __restrict__ V, int F, int H, int C) {
  int t = blockIdx.x * blockDim.x + threadIdx.x;
  if (t >= F * H) return;
  int f = t / H, h = t - f * H;
  const float* wr = W + (size_t)f * H * C + (size_t)h * C;
  const float* ar = avec + (size_t)h * C;
  float acc = 0.f;
  for (int c = 0; c < C; ++c) acc += wr[c] * ar[c];
  V[f * H + h] = acc;
}

// a[n,h] = sum_f X[n,f] * V[f,h]
__global__ void aproj_kernel(const float* __restrict__ X, const float* __restrict__ V,
                             float* __restrict__ a, long long NH, int F, int H) {
  long long t = (long long)blockIdx.x * blockDim.x + threadIdx.x;
  if (t >= NH) return;
  long long n = t / H;
  int h = (int)(t - n * H);
  const float* xr = X + n * (size_t)F;
  float acc = 0.f;
  for (int f = 0; f < F; ++f) acc += xr[f] * V[f * H + h];
  a[t] = acc;
}

// ---------------- WMMA f32 GEMM: Y[N,K] = X[N,F] @ W[F,K] ----------------
// Strip version: one wave owns a 16-row strip and computes all KTILES 16-col
// tiles, so each A fragment load feeds KTILES back-to-back WMMAs and X is
// read exactly once. Requires N%16==0, K==16*KTILES, F%4==0.
template <int KTILES>
__global__ void gemm16_wmma_f32_strip(const float* __restrict__ X,
                                      const float* __restrict__ W,
                                      float* __restrict__ Y, int N, int F, int K) {
  int strip = blockIdx.x * blockDim.y + threadIdx.y;   // uniform per wave
  if (strip >= (N >> 4)) return;                        // whole-wave exit, EXEC all-ones
  int lane = threadIdx.x;
  int r = lane & 15;
  int koff = (lane >> 4) << 1;       // lanes 0-15 -> K=0,1 ; lanes 16-31 -> K=2,3
  int m0 = strip << 4;
  v8f acc[KTILES];
#pragma unroll
  for (int t = 0; t < KTILES; ++t) acc[t] = (v8f){};
  const float* xr = X + (size_t)(m0 + r) * F + koff;
  const float* wr = W + (size_t)koff * K + r;
#if __has_builtin(__builtin_amdgcn_wmma_f32_16x16x4_f32)
  for (int k = 0; k < F; k += 4) {
    v2f a;
    a.x = xr[0];                     // A 16x4 fragment (row=r, K=koff/koff+1)
    a.y = xr[1];
    xr += 4;
#pragma unroll
    for (int t = 0; t < KTILES; ++t) {
      v2f b;
      b.x = wr[t * 16];              // B 4x16 fragment (K=koff/koff+1, col=r)
      b.y = wr[t * 16 + (size_t)K];
      acc[t] = __builtin_amdgcn_wmma_f32_16x16x4_f32(false, a, false, b, (short)0,
                                                     acc[t], false, false);
    }
    wr += (size_t)4 * K;
  }
#else
  for (int k = 0; k < F; ++k) {
    float a0 = X[(size_t)(m0 + ((lane >> 4) << 3)) * F];  // placeholder; fallback below
    (void)a0;
  }
#endif
  int mb = (lane >> 4) << 3;         // C/D: VGPR v holds rows mb+v
#if __has_builtin(__builtin_amdgcn_wmma_f32_16x16x4_f32)
#pragma unroll
  for (int t = 0; t < KTILES; ++t) {
    float* yr = Y + (size_t)(m0 + mb) * K + t * 16 + r;
#pragma unroll
    for (int v = 0; v < 8; ++v) yr[(size_t)v * K] = acc[t][v];
  }
#else
  // scalar fallback (same output mapping), only if the builtin is absent
#pragma unroll
  for (int t = 0; t < KTILES; ++t) {
    for (int v = 0; v < 8; ++v) {
      const float* xrow = X + (size_t)(m0 + mb + v) * F;
      const float* wcol = W + t * 16 + r;
      float s = 0.f;
      for (int k = 0; k < F; ++k) s += xrow[k] * wcol[(size_t)k * K];
      Y[(size_t)(m0 + mb + v) * K + t * 16 + r] = s;
    }
  }
#endif
}

// Generic single-tile version (fallback for shapes not covered by the strips).
__global__ void gemm16_wmma_f32(const float* __restrict__ X, const float* __restrict__ W,
                                float* __restrict__ Y, int N, int F, int K) {
  int ktiles = K >> 4;
  int tiles = (N >> 4) * ktiles;
  int tile = blockIdx.x * blockDim.y + threadIdx.y;
  if (tile >= tiles) return;
  int tm = tile / ktiles, tn = tile - tm * ktiles;
  int lane = threadIdx.x;
  int r = lane & 15;
  int koff = (lane >> 4) << 1;
  int m0 = tm << 4, n0 = tn << 4;
  const float* xr = X + (size_t)(m0 + r) * F + koff;
  const float* wr = W + (size_t)koff * K + (n0 + r);
  v8f c = {};
#if __has_builtin(__builtin_amdgcn_wmma_f32_16x16x4_f32)
  for (int k = 0; k < F; k += 4) {
    v2f a, b;
    a.x = xr[0];
    a.y = xr[1];
    b.x = wr[0];
    b.y = wr[(size_t)K];
    xr += 4;
    wr += (size_t)4 * K;
    c = __builtin_amdgcn_wmma_f32_16x16x4_f32(false, a, false, b, (short)0, c,
                                              false, false);
  }
  int mb = (lane >> 4) << 3;
  float* yr = Y + (size_t)(m0 + mb) * K + (n0 + r);
#pragma unroll
  for (int v = 0; v < 8; ++v) yr[(size_t)v * K] = c[v];
#else
  int mb = (lane >> 4) << 3;
#pragma unroll
  for (int v = 0; v < 8; ++v) {
    const float* xrow = X + (size_t)(m0 + mb + v) * F;
    const float* wcol = W + (n0 + r);
    float acc = 0.f;
    for (int k = 0; k < F; ++k) acc += xrow[k] * wcol[(size_t)k * K];
    Y[(size_t)(m0 + mb + v) * K + (n0 + r)] = acc;
  }
#endif
}

// ---------------- edge-wise segment softmax (3 passes) ----------------

__global__ void edge_max_kernel(const int* __restrict__ src, const int* __restrict__ dst,
                                long long EH, int H,
                                const float* __restrict__ a_src,
                                const float* __restrict__ a_dst,
                                unsigned* __restrict__ m) {
  long long idx = (long long)blockIdx.x * blockDim.x + threadIdx.x;
  if (idx >= EH) return;
  long long e = idx / H;
  int h = (int)(idx - e * H);
  int s = src[e], d = dst[e];
  float v = a_src[(size_t)s * H + h] + a_dst[(size_t)d * H + h];
  v = v > 0.f ? v : LRELU_SLOPE * v;
  atomicMax(m + (size_t)d * H + h, f2ord(v));
}

__global__ void edge_sum_kernel(const int* __restrict__ src, const int* __restrict__ dst,
                                long long EH, int H,
                                const float* __restrict__ a_src,
                                const float* __restrict__ a_dst,
                                const unsigned* __restrict__ m,
                                float* __restrict__ sums) {
  long long idx = (long long)blockIdx.x * blockDim.x + threadIdx.x;
  if (idx >= EH) return;
  long long e = idx / H;
  int h = (int)(idx - e * H);
  int s = src[e], d = dst[e];
  size_t off = (size_t)d * H + h;
  float v = a_src[(size_t)s * H + h] + a_dst[off];
  v = v > 0.f ? v : LRELU_SLOPE * v;
  atomicAdd(sums + off, __expf(v - ord2f(m[off])));
}

__global__ void edge_scatter_kernel(const int* __restrict__ src, const int* __restrict__ dst,
                                    long long EH, int H, int C,
                                    const float* __restrict__ a_src,
                                    const float* __restrict__ a_dst,
                                    const unsigned* __restrict__ m,
                                    const float* __restrict__ sums,
                                    const float* __restrict__ hs,
                                    float* __restrict__ out) {
  long long idx = (long long)blockIdx.x * blockDim.x + threadIdx.x;
  if (idx >= EH) return;
  long long e = idx / H;
  int h = (int)(idx - e * H);
  int s = src[e], d = dst[e];
  size_t off = (size_t)d * H + h;
  float v = a_src[(size_t)s * H + h] + a_dst[off];
  v = v > 0.f ? v : LRELU_SLOPE * v;
  float alpha = __expf(v - ord2f(m[off])) / sums[off];
  int HC = H * C;
  const float4* hrow = (const float4*)(hs + (size_t)s * HC + (size_t)h * C);
  float* orow = out + (size_t)d * HC + (size_t)h * C;
  int c4n = C >> 2;
  for (int c4 = 0; c4 < c4n; ++c4) {
    float4 hv = hrow[c4];
    atomicAdd(orow + c4 * 4 + 0, alpha * hv.x);
    atomicAdd(orow + c4 * 4 + 1, alpha * hv.y);
    atomicAdd(orow + c4 * 4 + 2, alpha * hv.z);
    atomicAdd(orow + c4 * 4 + 3, alpha * hv.w);
  }
}

// x = elu(scale * x) in place
__global__ void elu_scale_kernel(float* __restrict__ x, long long n, float scale) {
  long long i = (long long)blockIdx.x * blockDim.x + threadIdx.x;
  if (i < n) {
    float v = x[i] * scale;
    x[i] = v > 0.f ? v : (__expf(v) - 1.f);
  }
}

// out[0..31] = mean_i 0.5*o_svc[i,ch] ; out[32..63] = mean_i o_node[i,ch-32]
__global__ void pool_mean_kernel(const float* __restrict__ o_svc, int n_svc,
                                 const float* __restrict__ o_node, int n_node,
                                 float* __restrict__ out) {
  __shared__ float red[256];
  int ch = blockIdx.x;
  const float* srcp;
  int n, c;
  float scale;
  if (ch < 32) { srcp = o_svc; n = n_svc; c = ch; scale = 0.5f; }
  else         { srcp = o_node; n = n_node; c = ch - 32; scale = 1.0f; }
  float acc = 0.f;
  for (int i = threadIdx.x; i < n; i += blockDim.x) acc += srcp[(size_t)i * 32 + c];
  red[threadIdx.x] = acc;
  __syncthreads();
  for (int s2 = 128; s2 > 0; s2 >>= 1) {
    if ((int)threadIdx.x < s2) red[threadIdx.x] += red[threadIdx.x + s2];
    __syncthreads();
  }
  if (threadIdx.x == 0) out[ch] = red[0] * scale / (float)n;
}

// ---------------- host orchestration ----------------

struct RelP { const float *Ws, *Wd, *as, *ad, *b; };

static inline int NB(long long n, int t) { return (int)((n + t - 1) / t); }
static unsigned host_f2ord(float f) {
  unsigned u; memcpy(&u, &f, 4);
  return (u & 0x80000000u) ? ~u : (u | 0x80000000u);
}

static void launch_gemm(hipStream_t stream, const float* X, const float* W, float* Y,
                        int N, int F, int K) {
  dim3 blk(32, 4);
  if (K == 128) {
    long long strips = N / 16;
    gemm16_wmma_f32_strip<8><<<NB(strips, 4), blk, 0, stream>>>(X, W, Y, N, F, K);
  } else if (K == 32) {
    long long strips = N / 16;
    gemm16_wmma_f32_strip<2><<<NB(strips, 4), blk, 0, stream>>>(X, W, Y, N, F, K);
  } else {
    long long tiles = (long long)(N / 16) * (K / 16);
    gemm16_wmma_f32<<<NB(tiles, 4), blk, 0, stream>>>(X, W, Y, N, F, K);
  }
}

static void run_gat_rel(hipStream_t stream,
                        const float* x_src, int n_src, int f_src,
                        const float* x_dst, int n_dst, int f_dst,
                        const int* ei, long long E,
                        const RelP& p, int H, int C,
                        float* hs, float* a_src, float* a_dst,
                        unsigned* mbuf, float* sbuf, float* Vs, float* Vd,
                        float* out_acc) {
  if (E <= 0) return;
  int HC = H * C;
  vfold_kernel<<<NB((long long)f_src * H, 128), 128, 0, stream>>>(p.Ws, p.as, Vs, f_src, H, C);
  vfold_kernel<<<NB((long long)f_dst * H, 128), 128, 0, stream>>>(p.Wd, p.ad, Vd, f_dst, H, C);
  aproj_kernel<<<NB((long long)n_src * H, 256), 256, 0, stream>>>(x_src, Vs, a_src,
                                                                  (long long)n_src * H, f_src, H);
  aproj_kernel<<<NB((long long)n_dst * H, 256), 256, 0, stream>>>(x_dst, Vd, a_dst,
                                                                  (long long)n_dst * H, f_dst, H);
  launch_gemm(stream, x_src, p.Ws, hs, n_src, f_src, HC);
  long long ndh = (long long)n_dst * H;
  fill_u32_kernel<<<NB(ndh, 256), 256, 0, stream>>>(mbuf, ndh, host_f2ord(-3.0e38f));
  fill_u32_kernel<<<NB(ndh, 256), 256, 0, stream>>>((unsigned*)sbuf, ndh, 0u);
  long long EH = E * H;
  const int* srcp = ei;
  const int* dstp = ei + E;
  edge_max_kernel<<<NB(EH, 256), 256, 0, stream>>>(srcp, dstp, EH, H, a_src, a_dst, mbuf);
  edge_sum_kernel<<<NB(EH, 256), 256, 0, stream>>>(srcp, dstp, EH, H, a_src, a_dst, mbuf, sbuf);
  edge_scatter_kernel<<<NB(EH, 256), 256, 0, stream>>>(srcp, dstp, EH, H, C, a_src, a_dst,
                                                       mbuf, sbuf, hs, out_acc);
}

extern "C" void kernel_launch(void* const* d_in, const int* in_sizes, int n_in,
                              void* d_out, int out_size, void* d_ws, size_t ws_size,
                              hipStream_t stream) {
  const float* x_svc = (const float*)d_in[0];
  const float* x_node = (const float*)d_in[1];
  const int* ei_ss = (const int*)d_in[2];
  const int* ei_sn = (const int*)d_in[3];
  const int* ei_ns = (const int*)d_in[4];
  long long E_ss = in_sizes[2] / 2;
  long long E_sn = in_sizes[3] / 2;
  long long E_ns = in_sizes[4] / 2;
  const int F_SVC = 64, F_NODE = 32, HC0 = 128, OUTC = 32;
  int n_svc = in_sizes[0] / F_SVC;
  int n_node = in_sizes[1] / F_NODE;

  // params flattened leaf-by-leaf in dict order: per relation Ws, Wd, as, ad, b
  auto rel = [&](int base) -> RelP {
    return RelP{(const float*)d_in[base + 0], (const float*)d_in[base + 1],
                (const float*)d_in[base + 2], (const float*)d_in[base + 3],
                (const float*)d_in[base + 4]};
  };
  RelP l0_ss = rel(5), l0_sn = rel(10), l0_ns = rel(15);
  RelP l1_ss = rel(20), l1_sn = rel(25), l1_ns = rel(30);

  // workspace layout (floats)
  float* ws = (float*)d_ws;
  size_t off = 0;
  float* h_svc = ws + off;  off += (size_t)n_svc * HC0;   // layer-0 svc accumulator / layer-1 input
  float* h_node = ws + off; off += (size_t)n_node * HC0;
  float* hs = ws + off;     off += (size_t)n_svc * HC0;   // per-relation projected src (max size)
  float* a_src = ws + off;  off += (size_t)n_svc * 4;
  float* a_dst = ws + off;  off += (size_t)n_svc * 4;
  unsigned* mbuf = (unsigned*)(ws + off); off += (size_t)n_svc * 4;
  float* sbuf = ws + off;   off += (size_t)n_svc * 4;
  float* Vs = ws + off;     off += 1024;
  float* Vd = ws + off;     off += 1024;
  float* o_svc = ws + off;  off += (size_t)n_svc * OUTC;
  float* o_node = ws + off; off += (size_t)n_node * OUTC;
  (void)ws_size; (void)n_in; (void)out_size;

  // ---------- layer 0 (H=4, C=32, concat -> 128) ----------
  init_bias_kernel<<<NB((long long)n_svc * HC0, 256), 256, 0, stream>>>(
      h_svc, (long long)n_svc * HC0, HC0, l0_ss.b, l0_ns.b);
  init_bias_kernel<<<NB((long long)n_node * HC0, 256), 256, 0, stream>>>(
      h_node, (long long)n_node * HC0, HC0, l0_sn.b, nullptr);

  run_gat_rel(stream, x_svc, n_svc, F_SVC, x_svc, n_svc, F_SVC, ei_ss, E_ss,
              l0_ss, 4, 32, hs, a_src, a_dst, mbuf, sbuf, Vs, Vd, h_svc);
  run_gat_rel(stream, x_node, n_node, F_NODE, x_svc, n_svc, F_SVC, ei_ns, E_ns,
              l0_ns, 4, 32, hs, a_src, a_dst, mbuf, sbuf, Vs, Vd, h_svc);
  run_gat_rel(stream, x_svc, n_svc, F_SVC, x_node, n_node, F_NODE, ei_sn, E_sn,
              l0_sn, 4, 32, hs, a_src, a_dst, mbuf, sbuf, Vs, Vd, h_node);

  elu_scale_kernel<<<NB((long long)n_svc * HC0, 256), 256, 0, stream>>>(
      h_svc, (long long)n_svc * HC0, 0.5f);
  elu_scale_kernel<<<NB((long long)n_node * HC0, 256), 256, 0, stream>>>(
      h_node, (long long)n_node * HC0, 1.0f);

  // ---------- layer 1 (H=1, C=32) ----------
  init_bias_kernel<<<NB((long long)n_svc * OUTC, 256), 256, 0, stream>>>(
      o_svc, (long long)n_svc * OUTC, OUTC, l1_ss.b, l1_ns.b);
  init_bias_kernel<<<NB((long long)n_node * OUTC, 256), 256, 0, stream>>>(
      o_node, (long long)n_node * OUTC, OUTC, l1_sn.b, nullptr);

  run_gat_rel(stream, h_svc, n_svc, HC0, h_svc, n_svc, HC0, ei_ss, E_ss,
              l1_ss, 1, 32, hs, a_src, a_dst, mbuf, sbuf, Vs, Vd, o_svc);
  run_gat_rel(stream, h_node, n_node, HC0, h_svc, n_svc, HC0, ei_ns, E_ns,
              l1_ns, 1, 32, hs, a_src, a_dst, mbuf, sbuf, Vs, Vd, o_svc);
  run_gat_rel(stream, h_svc, n_svc, HC0, h_node, n_node, HC0, ei_sn, E_sn,
              l1_sn, 1, 32, hs, a_src, a_dst, mbuf, sbuf, Vs, Vd, o_node);

  // ---------- mean-pool + concat -> d_out[64] ----------
  pool_mean_kernel<<<64, 256, 0, stream>>>(o_svc, n_svc, o_node, n_node, (float*)d_out);
}